// RGCN_39384850104534
// MI455X (gfx1250) — compile-verified
//
#include <hip/hip_runtime.h>
#include <cstdint>

// ---------------------------------------------------------------------------
// RGCN (2-layer, basis decomposition) for MI455X / gfx1250.
//
// Restructure:  msg_e = sum_b att[et_e,b] * (x[src_e] @ basis_b)
//  => per layer:  H = X @ [basis0|basis1|basis2|basis3|root]   (one fp32 WMMA GEMM)
//     per edge:   msg = norm_e * sum_b att[et_e,b] * H[src_e, b*100:(b+1)*100]
//     scatter-add msg into agg[dst], count deg[dst]
//     out = agg/max(deg,1) + H[:,400:500] + bias   (+ ReLU for layer 1)
// ---------------------------------------------------------------------------

#define DD      100
#define NPAD    512        // 4*100 basis cols + 100 root cols + 12 zero pad
#define NNODES  50000
#define NEDGES  400000
#define KDIM    100

typedef __attribute__((ext_vector_type(2))) float v2f;
typedef __attribute__((ext_vector_type(8))) float v8f;

// Pack W (KDIM x NPAD):  cols [0,400) = basis[b][k][o],  [400,500) = root[k][o], rest 0
__global__ __launch_bounds__(256) void pack_w(const float* __restrict__ basis,
                                              const float* __restrict__ root,
                                              float* __restrict__ W) {
    int i = blockIdx.x * 256 + threadIdx.x;
    if (i >= KDIM * NPAD) return;
    int k = i / NPAD, c = i - k * NPAD;
    float v = 0.0f;
    if (c < 4 * DD)          v = basis[(c / DD) * (DD * DD) + k * DD + (c % DD)];
    else if (c < 5 * DD)     v = root[k * DD + (c - 4 * DD)];
    W[i] = v;
}

__global__ __launch_bounds__(256) void zero_f32(float* __restrict__ p, int n) {
    int i = blockIdx.x * 256 + threadIdx.x;
    if (i < n) p[i] = 0.0f;
}

// C(16 x 512) += A(16 x 100) * W(100 x 512), fp32 WMMA 16x16x4.
// One block per 16-row tile; 8 waves x (16x64) each; K stepped by 4.
__global__ __launch_bounds__(256) void gemm_wmma(const float* __restrict__ X, int ldx,
                                                 const float* __restrict__ W,
                                                 float* __restrict__ H) {
    __shared__ float As[16 * KDIM];
    const int m0  = blockIdx.x * 16;
    const int tid = threadIdx.x;

    for (int i = tid; i < 16 * KDIM; i += 256) {
        int r = i / KDIM, c = i - r * KDIM;
        As[i] = X[(size_t)(m0 + r) * ldx + c];
    }
    __syncthreads();

    const int lane    = tid & 31;
    const int wave    = tid >> 5;
    const int ln      = lane & 15;
    const int kh      = (lane >> 4) << 1;      // 0 for lanes 0-15, 2 for lanes 16-31
    const int colbase = wave * 64 + ln;        // wave covers cols [wave*64, wave*64+64)

    v8f acc0 = {}, acc1 = {}, acc2 = {}, acc3 = {};

    for (int k0 = 0; k0 < KDIM; k0 += 4) {
        // A fragment (ISA 7.12.2): lanes 0-15 -> row ln, K = k0,k0+1 ; lanes 16-31 -> K = k0+2,k0+3
        v2f a;
        a.x = As[ln * KDIM + k0 + kh];
        a.y = As[ln * KDIM + k0 + kh + 1];

        const float* Wk = W + (size_t)(k0 + kh) * NPAD + colbase;
        v2f b0; b0.x = Wk[0];  b0.y = Wk[NPAD];
        v2f b1; b1.x = Wk[16]; b1.y = Wk[NPAD + 16];
        v2f b2; b2.x = Wk[32]; b2.y = Wk[NPAD + 32];
        v2f b3; b3.x = Wk[48]; b3.y = Wk[NPAD + 48];

        acc0 = __builtin_amdgcn_wmma_f32_16x16x4_f32(false, a, false, b0, (short)0, acc0, false, false);
        acc1 = __builtin_amdgcn_wmma_f32_16x16x4_f32(false, a, false, b1, (short)0, acc1, false, false);
        acc2 = __builtin_amdgcn_wmma_f32_16x16x4_f32(false, a, false, b2, (short)0, acc2, false, false);
        acc3 = __builtin_amdgcn_wmma_f32_16x16x4_f32(false, a, false, b3, (short)0, acc3, false, false);
    }

    // C/D layout: VGPR j -> row j (lanes 0-15) or row j+8 (lanes 16-31), col = ln
    const int rbase = m0 + ((lane >> 4) << 3);
    float* Hp = H + (size_t)rbase * NPAD + colbase;
    #pragma unroll
    for (int j = 0; j < 8; ++j) {
        Hp[(size_t)j * NPAD + 0]  = acc0[j];
        Hp[(size_t)j * NPAD + 16] = acc1[j];
        Hp[(size_t)j * NPAD + 32] = acc2[j];
        Hp[(size_t)j * NPAD + 48] = acc3[j];
    }
}

// One wave32 per edge: gather 4x100 floats of H[src] (L2-resident), combine with
// att[et]*norm, scatter-add into agg[dst], bump deg[dst].
__global__ __launch_bounds__(256) void edge_phase(const float* __restrict__ H,
                                                  const int* __restrict__ ei,
                                                  const int* __restrict__ et,
                                                  const float* __restrict__ att,
                                                  const float* __restrict__ enorm,
                                                  float* __restrict__ agg,
                                                  float* __restrict__ deg) {
    int g    = blockIdx.x * 256 + threadIdx.x;
    int e    = g >> 5;
    int lane = g & 31;
    if (e >= NEDGES) return;

    int   src = ei[e];
    int   dst = ei[NEDGES + e];
    int   r   = et[e];
    float nm  = enorm[e];
    float a0 = att[r * 4 + 0] * nm;
    float a1 = att[r * 4 + 1] * nm;
    float a2 = att[r * 4 + 2] * nm;
    float a3 = att[r * 4 + 3] * nm;

    if (lane == 0) atomicAdd(deg + dst, 1.0f);

    if (lane < 25) {                              // 25 float4 = 100 floats
        const float4* Hr = (const float4*)(H + (size_t)src * NPAD);
        float4 h0 = Hr[lane];
        float4 h1 = Hr[25 + lane];
        float4 h2 = Hr[50 + lane];
        float4 h3 = Hr[75 + lane];
        float4 m;
        m.x = a0 * h0.x + a1 * h1.x + a2 * h2.x + a3 * h3.x;
        m.y = a0 * h0.y + a1 * h1.y + a2 * h2.y + a3 * h3.y;
        m.z = a0 * h0.z + a1 * h1.z + a2 * h2.z + a3 * h3.z;
        m.w = a0 * h0.w + a1 * h1.w + a2 * h2.w + a3 * h3.w;
        float* ap = agg + (size_t)dst * DD + lane * 4;
        atomicAdd(ap + 0, m.x);
        atomicAdd(ap + 1, m.y);
        atomicAdd(ap + 2, m.z);
        atomicAdd(ap + 3, m.w);
    }
}

__global__ __launch_bounds__(256) void finalize(const float* __restrict__ agg,
                                                const float* __restrict__ deg,
                                                const float* __restrict__ H,
                                                const float* __restrict__ bias,
                                                float* __restrict__ out, int relu) {
    int i = blockIdx.x * 256 + threadIdx.x;
    if (i >= NNODES * DD) return;
    int n = i / DD, d = i - n * DD;
    float v = agg[i] / fmaxf(deg[n], 1.0f) + H[(size_t)n * NPAD + 4 * DD + d] + bias[d];
    out[i] = relu ? fmaxf(v, 0.0f) : v;
}

extern "C" void kernel_launch(void* const* d_in, const int* in_sizes, int n_in,
                              void* d_out, int out_size, void* d_ws, size_t ws_size,
                              hipStream_t stream) {
    const float* entity = (const float*)d_in[0];
    const int*   ei     = (const int*)d_in[1];
    // d_in[2] edge_attr: unused by the reference
    const int*   et     = (const int*)d_in[3];
    const float* enorm  = (const float*)d_in[4];
    const float* basis1 = (const float*)d_in[5];
    const float* att1   = (const float*)d_in[6];
    const float* root1  = (const float*)d_in[7];
    const float* bias1  = (const float*)d_in[8];
    const float* basis2 = (const float*)d_in[9];
    const float* att2   = (const float*)d_in[10];
    const float* root2  = (const float*)d_in[11];
    const float* bias2  = (const float*)d_in[12];
    float* out = (float*)d_out;

    // Workspace layout (floats):
    float* H    = (float*)d_ws;                       // NNODES * NPAD   (102.4 MB)
    float* agg  = H    + (size_t)NNODES * NPAD;       // NNODES * DD     (20 MB)
    float* deg  = agg  + (size_t)NNODES * DD;         // NNODES          (0.2 MB), contiguous with agg
    float* hbuf = deg  + NNODES;                      // NNODES * DD     (20 MB)
    float* Wp   = hbuf + (size_t)NNODES * DD;         // KDIM * NPAD     (0.2 MB)

    dim3 blk(256);
    int packB = (KDIM * NPAD + 255) / 256;
    int zeroN = NNODES * DD + NNODES;                 // agg + deg contiguous
    int zeroB = (zeroN + 255) / 256;
    int gemmB = NNODES / 16;                          // 3125, exact
    int edgeB = (NEDGES * 32 + 255) / 256;            // one wave per edge
    int finB  = (NNODES * DD + 255) / 256;

    // ---- layer 1 ----
    pack_w    <<<packB, blk, 0, stream>>>(basis1, root1, Wp);
    zero_f32  <<<zeroB, blk, 0, stream>>>(agg, zeroN);
    gemm_wmma <<<gemmB, blk, 0, stream>>>(entity, DD, Wp, H);
    edge_phase<<<edgeB, blk, 0, stream>>>(H, ei, et, att1, enorm, agg, deg);
    finalize  <<<finB,  blk, 0, stream>>>(agg, deg, H, bias1, hbuf, 1);

    // ---- layer 2 ----
    pack_w    <<<packB, blk, 0, stream>>>(basis2, root2, Wp);
    zero_f32  <<<zeroB, blk, 0, stream>>>(agg, zeroN);
    gemm_wmma <<<gemmB, blk, 0, stream>>>(hbuf, DD, Wp, H);
    edge_phase<<<edgeB, blk, 0, stream>>>(H, ei, et, att2, enorm, agg, deg);
    finalize  <<<finB,  blk, 0, stream>>>(agg, deg, H, bias2, out, 0);
}